// QuadraticNet_50208167690683
// MI455X (gfx1250) — compile-verified
//
#include <hip/hip_runtime.h>
#include <cstddef>

typedef __attribute__((ext_vector_type(2))) float v2f;
typedef __attribute__((ext_vector_type(8))) float v8f;

#define DEV __device__ __forceinline__

DEV float eluf(float v) { return v > 0.f ? v : (__expf(v) - 1.f); }
DEV float histf(float v, float bin, float nm1) {
  return fmaxf(1.f - fabsf(v - bin) * nm1, 0.f);
}

// D = A(16x4 f32) * B(4x16 f32) + C(16x16 f32)
DEV v8f wmma_f32(v2f a, v2f b, v8f c) {
  return __builtin_amdgcn_wmma_f32_16x16x4_f32(false, a, false, b, (short)0, c,
                                               false, false);
}

// ---------------------------------------------------------------------------
// Weight transposes to K-major, pair-packed layout:
//   wT[((k>>1)*64 + n)*2 + (k&1)]  -> lane loads a float2 = (B[k],B[k+1])[n]
// conv0: k = (ky*3+kx)*96 + cf  (cf padded 93->96), K = 864
// conv1/2: k = (ky*3+kx)*64 + cin, K = 576
// ---------------------------------------------------------------------------
__global__ void tr0_kernel(const float* __restrict__ w, float* __restrict__ wT) {
  int idx = blockIdx.x * blockDim.x + threadIdx.x;
  if (idx >= 864 * 64) return;
  int k = idx >> 6, n = idx & 63;
  int r = k / 96, cf = k % 96;
  float v = (cf < 93) ? w[(size_t)n * 837 + cf * 9 + r] : 0.f;
  wT[((((size_t)(k >> 1)) * 64 + n) << 1) + (k & 1)] = v;
}

__global__ void tr1_kernel(const float* __restrict__ w, float* __restrict__ wT) {
  int idx = blockIdx.x * blockDim.x + threadIdx.x;
  if (idx >= 576 * 64) return;
  int k = idx >> 6, n = idx & 63;
  int r = k >> 6, cin = k & 63;
  float v = w[(size_t)n * 576 + cin * 9 + r];
  wT[((((size_t)(k >> 1)) * 64 + n) << 1) + (k & 1)] = v;
}

// ---------------------------------------------------------------------------
// conv0: fused feature synthesis + implicit GEMM + bias + ELU + 2x2 maxpool.
// One wave per workgroup. Tile: 16 conv-x x 2 conv-y x 64 channels
// -> pooled 8 x 1 x 64.  Grid: (16 x-tiles, 127 pooled-y, 4 batch).
//
// Phase 1: each of the 165 (= 5 rows x 33 cols) input pixels the tile touches
// gets its full 96-entry feature vector (x, sobel, hist4/8/16, zero-pad)
// computed once, branch-free, into LDS ft[pixel][98] (stride padded for
// bank-conflict-free ds_load_b64 across the 16 M-lanes).
// Phase 2: K-loop (K = 864) is pure ds_load_b64 (A) + global b64 (B) + WMMA.
// Output f1 channel-last (B,127,127,64).
// ---------------------------------------------------------------------------
__global__ __launch_bounds__(32) void conv0_kernel(const float* __restrict__ x,
                                                   const float* __restrict__ wT,
                                                   const float* __restrict__ bias,
                                                   float* __restrict__ f1) {
  const int lane = threadIdx.x;
  const int hh = lane >> 4;   // K-half (A/B) and M-high-half (C/D)
  const int ml = lane & 15;   // M lane for A, N lane for B/C/D
  const int tx = blockIdx.x;
  const int py = blockIdx.y;
  const int b = blockIdx.z;

  __shared__ float ft[165][98];  // [ly*33+lx][cf], 63.2 KiB

  const int gy0 = 4 * py;
  const int gx0 = 32 * tx;
  const float* xb = x + (size_t)b * 3 * 512 * 512;

  // ---- Phase 1: per-pixel feature synthesis (replicate padding via clamp) --
  for (int p = lane; p < 165; p += 32) {
    const int ly = p / 33, lx = p - ly * 33;
    const int gy = min(gy0 + ly, 511);
    const int gx = min(gx0 + lx, 511);
    const int ym = max(gy - 1, 0), yp = min(gy + 1, 511);
    const int xm = max(gx - 1, 0), xp = min(gx + 1, 511);
    float* f = &ft[p][0];
    float cv[3];
    int o = 9;
#pragma unroll
    for (int c = 0; c < 3; ++c) {
      const float* xc = xb + (size_t)c * 512 * 512;
      float a = xc[ym * 512 + xm], bb = xc[ym * 512 + gx], cc = xc[ym * 512 + xp];
      float d = xc[gy * 512 + xm], ee = xc[gy * 512 + gx], e = xc[gy * 512 + xp];
      float g = xc[yp * 512 + xm], h = xc[yp * 512 + gx], i = xc[yp * 512 + xp];
      cv[c] = ee;
      f[c] = ee;
      f[3 + 2 * c] = ((cc + 2.f * e + i) - (a + 2.f * d + g)) * 0.125f;  // dx
      f[4 + 2 * c] = ((g + 2.f * h + i) - (a + 2.f * bb + cc)) * 0.125f; // dy
    }
#pragma unroll
    for (int c = 0; c < 3; ++c)
#pragma unroll
      for (int bn = 0; bn < 4; ++bn) f[o++] = histf(cv[c], (float)bn * (1.f / 3.f), 3.f);
#pragma unroll
    for (int c = 0; c < 3; ++c)
#pragma unroll
      for (int bn = 0; bn < 8; ++bn) f[o++] = histf(cv[c], (float)bn * (1.f / 7.f), 7.f);
#pragma unroll
    for (int c = 0; c < 3; ++c)
#pragma unroll
      for (int bn = 0; bn < 16; ++bn) f[o++] = histf(cv[c], (float)bn * (1.f / 15.f), 15.f);
    f[93] = 0.f;
    f[94] = 0.f;
    f[95] = 0.f;
  }
  __syncthreads();

  // ---- Phase 2: implicit GEMM over K = 864 --------------------------------
  v8f acc[2][4];
  const v8f vz = {0.f, 0.f, 0.f, 0.f, 0.f, 0.f, 0.f, 0.f};
  for (int mt = 0; mt < 2; ++mt)
    for (int nt = 0; nt < 4; ++nt) acc[mt][nt] = vz;

  for (int r = 0; r < 9; ++r) {  // filter tap
    const int ky = r / 3, kx = r % 3;
    const int lx = 2 * ml + kx;
    const float* fp0 = &ft[ky * 33 + lx][2 * hh];        // conv row 2*py
    const float* fp1 = &ft[(ky + 2) * 33 + lx][2 * hh];  // conv row 2*py+1
    // prefetch next tap's B panel into L2 (global_prefetch_b8)
    __builtin_prefetch(wT + (((size_t)(min(r + 1, 8) * 48)) * 64) * 2, 0, 1);
    for (int u = 0; u < 24; ++u) {  // 96 / 4 K-steps within the tap
      v2f a0 = *(const v2f*)(fp0 + 4 * u);
      v2f a1 = *(const v2f*)(fp1 + 4 * u);
      const float* bp = wT + (((size_t)(r * 48 + 2 * u + hh)) * 64) * 2;
#pragma unroll
      for (int nt = 0; nt < 4; ++nt) {
        v2f bv = *(const v2f*)(bp + ((nt * 16 + ml) << 1));
        acc[0][nt] = wmma_f32(a0, bv, acc[0][nt]);
        acc[1][nt] = wmma_f32(a1, bv, acc[1][nt]);
      }
    }
  }

  // bias + ELU + 2x2 maxpool (M = vgpr + 8*hh, N = ml per C/D layout)
  const int pxb = 8 * tx;
#pragma unroll
  for (int nt = 0; nt < 4; ++nt) {
    const float bb = bias[nt * 16 + ml];
#pragma unroll
    for (int j = 0; j < 4; ++j) {
      float m0 = fmaxf(eluf(acc[0][nt][2 * j] + bb), eluf(acc[0][nt][2 * j + 1] + bb));
      float m1 = fmaxf(eluf(acc[1][nt][2 * j] + bb), eluf(acc[1][nt][2 * j + 1] + bb));
      int px = pxb + j + 4 * hh;
      if (px < 127)
        f1[((((size_t)b * 127 + py) * 127 + px) << 6) + nt * 16 + ml] = fmaxf(m0, m1);
    }
  }
}

// ---------------------------------------------------------------------------
// conv1 / conv2: 64->64 implicit GEMM (K = 576) + bias + ELU + 2x2 maxpool.
// Input channel-last (B,HIN,WIN,64); output channel-last (B,HP,WP,64).
// A fragments loaded directly from global (L2-resident), float2 per lane.
// ---------------------------------------------------------------------------
template <int HIN, int WIN, int HP, int WP>
__global__ __launch_bounds__(32) void convmid_kernel(const float* __restrict__ fin,
                                                     const float* __restrict__ wT,
                                                     const float* __restrict__ bias,
                                                     float* __restrict__ fout) {
  const int lane = threadIdx.x;
  const int hh = lane >> 4;
  const int ml = lane & 15;
  const int tx = blockIdx.x;
  const int py = blockIdx.y;
  const int b = blockIdx.z;

  v8f acc[2][4];
  const v8f vz = {0.f, 0.f, 0.f, 0.f, 0.f, 0.f, 0.f, 0.f};
  for (int mt = 0; mt < 2; ++mt)
    for (int nt = 0; nt < 4; ++nt) acc[mt][nt] = vz;

  const int cx = 16 * tx + ml;  // conv-x for this lane's M
  const float* finb = fin + (size_t)b * HIN * WIN * 64;

  for (int r = 0; r < 9; ++r) {
    const int ky = r / 3, kx = r % 3;
    const int xx = min(2 * cx + kx, WIN - 1);
    const int yy0 = min(4 * py + ky, HIN - 1);
    const int yy1 = min(4 * py + 2 + ky, HIN - 1);
    const float* row0 = finb + (((size_t)yy0 * WIN + xx) << 6);
    const float* row1 = finb + (((size_t)yy1 * WIN + xx) << 6);
    __builtin_prefetch(wT + (((size_t)(min(r + 1, 8) * 32)) * 64) * 2, 0, 1);
    for (int u = 0; u < 16; ++u) {  // 64 input channels / 4
      const int cinb = 4 * u + 2 * hh;
      v2f a0 = *(const v2f*)(row0 + cinb);
      v2f a1 = *(const v2f*)(row1 + cinb);
      const float* bp = wT + (((size_t)(r * 32 + 2 * u + hh)) * 64) * 2;
#pragma unroll
      for (int nt = 0; nt < 4; ++nt) {
        v2f bv = *(const v2f*)(bp + ((nt * 16 + ml) << 1));
        acc[0][nt] = wmma_f32(a0, bv, acc[0][nt]);
        acc[1][nt] = wmma_f32(a1, bv, acc[1][nt]);
      }
    }
  }

  const int pxb = 8 * tx;
#pragma unroll
  for (int nt = 0; nt < 4; ++nt) {
    const float bb = bias[nt * 16 + ml];
#pragma unroll
    for (int j = 0; j < 4; ++j) {
      float m0 = fmaxf(eluf(acc[0][nt][2 * j] + bb), eluf(acc[0][nt][2 * j + 1] + bb));
      float m1 = fmaxf(eluf(acc[1][nt][2 * j] + bb), eluf(acc[1][nt][2 * j + 1] + bb));
      int px = pxb + j + 4 * hh;
      if (px < WP)
        fout[((((size_t)b * HP + py) * WP + px) << 6) + nt * 16 + ml] = fmaxf(m0, m1);
    }
  }
}

// ---------------------------------------------------------------------------
// Global mean over 7x7 -> (B,64)
// ---------------------------------------------------------------------------
__global__ void mean_kernel(const float* __restrict__ f3, float* __restrict__ gm) {
  int b = blockIdx.x, c = threadIdx.x;  // 64 threads
  float s = 0.f;
  for (int i = 0; i < 49; ++i) s += f3[(((size_t)b * 49 + i) << 6) + c];
  gm[b * 64 + c] = s * (1.f / 49.f);
}

// ---------------------------------------------------------------------------
// fc1 (64x64)+ELU -> fc2 (30x64): per-image quadratic coefficients (B,30)
// ---------------------------------------------------------------------------
__global__ void fc_kernel(const float* __restrict__ gm, const float* __restrict__ w1,
                          const float* __restrict__ b1, const float* __restrict__ w2,
                          const float* __restrict__ b2, float* __restrict__ wq) {
  __shared__ float fv[64];
  __shared__ float h[64];
  int b = blockIdx.x, j = threadIdx.x;  // 64 threads
  fv[j] = gm[b * 64 + j];
  __syncthreads();
  float acc = b1[j];
  for (int c = 0; c < 64; ++c) acc += fv[c] * w1[j * 64 + c];
  h[j] = eluf(acc);
  __syncthreads();
  if (j < 30) {
    float a2 = b2[j];
    for (int c = 0; c < 64; ++c) a2 += h[c] * w2[j * 64 + c];
    wq[b * 30 + j] = a2;
  }
}

// ---------------------------------------------------------------------------
// y = x + quadratic(x, w):   feats = upper-tri of (r,g,b,1) outer product,
// out_c = x_c + sum_k feats[k] * wq[b][c*10 + k]
// ---------------------------------------------------------------------------
__global__ void quad_kernel(const float* __restrict__ x, const float* __restrict__ wq,
                            float* __restrict__ out) {
  const int HW = 512 * 512;
  int idx = blockIdx.x * blockDim.x + threadIdx.x;
  if (idx >= 4 * HW) return;
  int b = idx / HW, hw = idx - b * HW;
  const float* xb = x + (size_t)b * 3 * HW;
  float r = xb[hw], g = xb[HW + hw], bl = xb[2 * HW + hw];
  float f[10] = {r * r, r * g, r * bl, r, g * g, g * bl, g, bl * bl, bl, 1.f};
  const float* w = wq + b * 30;
  float* ob = out + (size_t)b * 3 * HW;
#pragma unroll
  for (int c = 0; c < 3; ++c) {
    float s = (c == 0) ? r : ((c == 1) ? g : bl);
#pragma unroll
    for (int k = 0; k < 10; ++k) s += f[k] * w[c * 10 + k];
    ob[c * HW + hw] = s;
  }
}

// ---------------------------------------------------------------------------
// Workspace layout (float offsets); total ~18.1 MB
// ---------------------------------------------------------------------------
static const size_t OFF_WT0 = 0;                        // 864*64    = 55296
static const size_t OFF_WT1 = 55296;                    // 576*64    = 36864
static const size_t OFF_WT2 = 92160;                    // 576*64    = 36864
static const size_t OFF_F1 = 129024;                    // 4*127*127*64
static const size_t OFF_F2 = 129024 + 4129024;          // 4*31*31*64
static const size_t OFF_F3 = 129024 + 4129024 + 246016; // 4*7*7*64
static const size_t OFF_GM = OFF_F3 + 12544;            // 4*64
static const size_t OFF_WQ = OFF_GM + 256;              // 4*30

extern "C" void kernel_launch(void* const* d_in, const int* in_sizes, int n_in,
                              void* d_out, int out_size, void* d_ws, size_t ws_size,
                              hipStream_t stream) {
  const float* x = (const float*)d_in[0];
  const float* c0w = (const float*)d_in[1];
  const float* c0b = (const float*)d_in[2];
  const float* c1w = (const float*)d_in[3];
  const float* c1b = (const float*)d_in[4];
  const float* c2w = (const float*)d_in[5];
  const float* c2b = (const float*)d_in[6];
  const float* f1w = (const float*)d_in[7];
  const float* f1b = (const float*)d_in[8];
  const float* f2w = (const float*)d_in[9];
  const float* f2b = (const float*)d_in[10];
  float* out = (float*)d_out;
  float* ws = (float*)d_ws;

  float* wT0 = ws + OFF_WT0;
  float* wT1 = ws + OFF_WT1;
  float* wT2 = ws + OFF_WT2;
  float* F1 = ws + OFF_F1;
  float* F2 = ws + OFF_F2;
  float* F3 = ws + OFF_F3;
  float* GM = ws + OFF_GM;
  float* WQ = ws + OFF_WQ;

  tr0_kernel<<<(864 * 64 + 255) / 256, 256, 0, stream>>>(c0w, wT0);
  tr1_kernel<<<(576 * 64 + 255) / 256, 256, 0, stream>>>(c1w, wT1);
  tr1_kernel<<<(576 * 64 + 255) / 256, 256, 0, stream>>>(c2w, wT2);

  conv0_kernel<<<dim3(16, 127, 4), 32, 0, stream>>>(x, wT0, c0b, F1);
  convmid_kernel<127, 127, 31, 31><<<dim3(4, 31, 4), 32, 0, stream>>>(F1, wT1, c1b, F2);
  convmid_kernel<31, 31, 7, 7><<<dim3(1, 7, 4), 32, 0, stream>>>(F2, wT2, c2b, F3);

  mean_kernel<<<4, 64, 0, stream>>>(F3, GM);
  fc_kernel<<<4, 64, 0, stream>>>(GM, f1w, f1b, f2w, f2b, WQ);

  const int npix = 4 * 512 * 512;
  quad_kernel<<<(npix + 255) / 256, 256, 0, stream>>>(x, WQ, out);
}